// EVI_6794638262475
// MI455X (gfx1250) — compile-verified
//
#include <hip/hip_runtime.h>

typedef float v2f __attribute__((ext_vector_type(2)));
typedef float v8f __attribute__((ext_vector_type(8)));

#define HWD    262144      // 64*64*64 spatial positions per batch
#define NCH    64
#define NCLS   4
#define NPROTO 20
#define NVOX   524288      // 2 * HWD
#define NCHUNK 2048        // NVOX / 256
#define PSTR   12          // padded proto-table stride (floats), 48B -> b128-aligned

__global__ __launch_bounds__(256)
void evi_fused_kernel(const float* __restrict__ x,
                      const float* __restrict__ w_to,
                      const float* __restrict__ b_to,
                      const float* __restrict__ w_from,
                      const float* __restrict__ b_from,
                      const float* __restrict__ Wp,
                      const float* __restrict__ BETA,
                      const float* __restrict__ alpha,
                      const float* __restrict__ gam,
                      float* __restrict__ out)
{
    // Per-prototype table: [k*12 + {W0..W3, U0..U3, alphap, -0.5*g2, pad, pad}]
    __shared__ float sP[NPROTO * PSTR];

    const int tid = threadIdx.x;
    if (tid < NPROTO) {
        float b0 = BETA[tid*4+0], b1 = BETA[tid*4+1];
        float b2 = BETA[tid*4+2], b3 = BETA[tid*4+3];
        float u0 = b0*b0, u1 = b1*b1, u2 = b2*b2, u3 = b3*b3;
        float inv = 1.0f / (u0 + u1 + u2 + u3);
        sP[tid*PSTR+0] = Wp[tid*4+0];
        sP[tid*PSTR+1] = Wp[tid*4+1];
        sP[tid*PSTR+2] = Wp[tid*4+2];
        sP[tid*PSTR+3] = Wp[tid*4+3];
        sP[tid*PSTR+4] = u0 * inv;
        sP[tid*PSTR+5] = u1 * inv;
        sP[tid*PSTR+6] = u2 * inv;
        sP[tid*PSTR+7] = u3 * inv;
        sP[tid*PSTR+8] = 0.99f / (1.0f + __expf(-alpha[tid]));
        float g = gam[tid];
        sP[tid*PSTR+9]  = -0.5f * g * g;
        sP[tid*PSTR+10] = 0.0f;
        sP[tid*PSTR+11] = 0.0f;
    }
    __syncthreads();

    const int lane  = tid & 31;
    const int wave  = tid >> 5;
    const int m     = lane & 15;       // row within A / voxel within tile
    const int khalf = lane >> 4;       // which K-pair this lane supplies
    const int cls   = m & 3;
    const bool avalid = ((m >> 2) & 1) == 0;   // rows 0-3 and 8-11 carry w_to (dup)

    // ---- A operand for projection 1: w_to, classes on rows 0-3 and 8-11 ----
    v2f a1[16];
#pragma unroll
    for (int j = 0; j < 16; ++j) {
        int c0 = 4*j + 2*khalf;
        a1[j].x = avalid ? w_to[cls*NCH + c0]     : 0.0f;
        a1[j].y = avalid ? w_to[cls*NCH + c0 + 1] : 0.0f;
    }
    // ---- A operand for projection 2: w_from, 4 groups of 16 channel rows ----
    v2f a2[4];
#pragma unroll
    for (int g = 0; g < 4; ++g) {
        a2[g].x = w_from[(16*g + m)*NCLS + 2*khalf];
        a2[g].y = w_from[(16*g + m)*NCLS + 2*khalf + 1];
    }
    // ---- C-init for projection 2: b_from broadcast in C/D layout ----
    v8f c2init[4];
#pragma unroll
    for (int g = 0; g < 4; ++g) {
#pragma unroll
        for (int r = 0; r < 8; ++r)
            c2init[g][r] = b_from[16*g + 8*khalf + r];
    }

    const float bt0 = b_to[0], bt1 = b_to[1], bt2 = b_to[2], bt3 = b_to[3];

    for (int chunk = blockIdx.x; chunk < NCHUNK; chunk += gridDim.x) {
        const uint32_t vbase = (uint32_t)chunk * 256u + (uint32_t)wave * 32u;
        const uint32_t batch = vbase >> 18;               // HWD = 2^18
        const uint32_t s0    = vbase & (HWD - 1u);
        // 32-bit element offset of (batch, channel 2*khalf, voxel s0+m)
        const uint32_t pbase = (batch << 24)              // batch * NCH * HWD
                             + (uint32_t)(2*khalf) * HWD + s0 + (uint32_t)m;

        // ================= projection 1: ev = w_to * x^T ====================
        v8f d0 = {0.f,0.f,0.f,0.f,0.f,0.f,0.f,0.f};
        v8f d1 = {0.f,0.f,0.f,0.f,0.f,0.f,0.f,0.f};
#pragma unroll
        for (int j = 0; j < 16; ++j) {
            const uint32_t p = pbase + (uint32_t)(4*j) * HWD;
            v2f bv0, bv1;
            bv0.x = __builtin_nontemporal_load(x + p);
            bv0.y = __builtin_nontemporal_load(x + p + HWD);
            bv1.x = __builtin_nontemporal_load(x + p + 16u);
            bv1.y = __builtin_nontemporal_load(x + p + HWD + 16u);
            d0 = __builtin_amdgcn_wmma_f32_16x16x4_f32(false, a1[j], false, bv0,
                                                       (short)0, d0, false, false);
            d1 = __builtin_amdgcn_wmma_f32_16x16x4_f32(false, a1[j], false, bv1,
                                                       (short)0, d1, false, false);
        }
        // Each lane picks up ev[0..3] for voxel (s0 + lane):
        // lanes 0-15 from d0 rows 0-3, lanes 16-31 from d1 rows 8-11 (dup rows).
        const bool lo = (lane < 16);
        float ev0 = (lo ? d0[0] : d1[0]) + bt0;
        float ev1 = (lo ? d0[1] : d1[1]) + bt1;
        float ev2 = (lo ? d0[2] : d1[2]) + bt2;
        float ev3 = (lo ? d0[3] : d1[3]) + bt3;

        // ================= Dempster-Shafer scan over 20 prototypes ==========
        // Serially dependent: keep rolled to bound live ranges / VGPR count.
        float mk0 = 0.f, mk1 = 0.f, mk2 = 0.f, mk3 = 0.f, mk5 = 1.f;
#pragma unroll 1
        for (int k = 0; k < NPROTO; ++k) {
            const float* pk = &sP[k*PSTR];
            float q0 = ev0 - pk[0], q1 = ev1 - pk[1];
            float q2 = ev2 - pk[2], q3 = ev3 - pk[3];
            float dd = q0*q0 + q1*q1 + q2*q2 + q3*q3;
            float s  = pk[8] * __expf(pk[9] * dd);   // ap * exp(-0.5*g2*dd)
            float m5 = 1.0f - s;
            float mc0 = pk[4]*s, mc1 = pk[5]*s, mc2 = pk[6]*s, mc3 = pk[7]*s;
            mk0 = mk0*(mc0 + m5) + mc0*mk5;
            mk1 = mk1*(mc1 + m5) + mc1*mk5;
            mk2 = mk2*(mc2 + m5) + mc2*mk5;
            mk3 = mk3*(mc3 + m5) + mc3*mk5;
            mk5 = mk5 * m5;
        }
        float Ksum = mk0 + mk1 + mk2 + mk3 + mk5;
        float rinv = __builtin_amdgcn_rcpf(Ksum);
        float e0 = (mk0 + mk5) * rinv;
        float e1 = (mk1 + mk5) * rinv;
        float e2 = (mk2 + mk5) * rinv;
        float e3 = (mk3 + mk5) * rinv;

        // ================= projection 2: out = w_from * evid + b_from =======
#pragma unroll
        for (int t = 0; t < 2; ++t) {
            // Gather evid into WMMA B layout from the lanes owning tile t
            const int idx = ((t << 4) + m) << 2;     // byte lane index
            float g0 = __int_as_float(__builtin_amdgcn_ds_bpermute(idx, __float_as_int(e0)));
            float g1 = __int_as_float(__builtin_amdgcn_ds_bpermute(idx, __float_as_int(e1)));
            float g2 = __int_as_float(__builtin_amdgcn_ds_bpermute(idx, __float_as_int(e2)));
            float g3 = __int_as_float(__builtin_amdgcn_ds_bpermute(idx, __float_as_int(e3)));
            v2f b2v;
            b2v.x = khalf ? g2 : g0;                 // K = 2*khalf
            b2v.y = khalf ? g3 : g1;                 // K = 2*khalf + 1
            const uint32_t sbase = s0 + (uint32_t)(16*t + m);
#pragma unroll
            for (int g = 0; g < 4; ++g) {
                v8f acc = __builtin_amdgcn_wmma_f32_16x16x4_f32(false, a2[g], false, b2v,
                                                                (short)0, c2init[g],
                                                                false, false);
                const uint32_t obase = (batch << 24)
                                     + (uint32_t)(16*g + 8*khalf) * HWD + sbase;
#pragma unroll
                for (int r = 0; r < 8; ++r)
                    __builtin_nontemporal_store(acc[r], out + obase + (uint32_t)r * HWD);
            }
        }
    }
}

extern "C" void kernel_launch(void* const* d_in, const int* in_sizes, int n_in,
                              void* d_out, int out_size, void* d_ws, size_t ws_size,
                              hipStream_t stream) {
    const float* x      = (const float*)d_in[0];
    const float* w_to   = (const float*)d_in[1];
    const float* b_to   = (const float*)d_in[2];
    const float* w_from = (const float*)d_in[3];
    const float* b_from = (const float*)d_in[4];
    const float* Wp     = (const float*)d_in[5];
    const float* BETA   = (const float*)d_in[6];
    const float* alpha  = (const float*)d_in[7];
    const float* gam    = (const float*)d_in[8];
    float* out = (float*)d_out;

    evi_fused_kernel<<<1024, 256, 0, stream>>>(x, w_to, b_to, w_from, b_from,
                                               Wp, BETA, alpha, gam, out);
}